// EKG_15315853377999
// MI455X (gfx1250) — compile-verified
//
#include <hip/hip_runtime.h>
#include <hip/hip_bf16.h>

typedef __attribute__((ext_vector_type(2))) float v2f;
typedef __attribute__((ext_vector_type(8))) float v8f;
typedef int v4i_vs __attribute__((vector_size(4 * sizeof(int))));

#define GLOBAL_AS __attribute__((address_space(1)))
#define LDS_AS    __attribute__((address_space(3)))

enum : int { BATCH = 16, CIN = 64, HH = 256, WW = 256, CKK = 576, FCD = 512, TILE_H = 8 };

#ifndef __has_builtin
#define __has_builtin(x) 0
#endif

#if __has_builtin(__builtin_amdgcn_global_load_async_to_lds_b128)
#define HAVE_ASYNC_LDS 1
#else
#define HAVE_ASYNC_LDS 0
#endif

// ---------------------------------------------------------------------------
// MLP GEMM via V_WMMA_F32_16X16X4_F32: out(16 x 576) = A(16 x K) @ Bm(K x 576)
// One wave (32 threads) per 16-column tile. 36 tiles.
// A layout (32-bit 16x4): lanes 0-15 -> M=lane, VGPR0=K0,VGPR1=K1;
//                         lanes 16-31 -> M=lane-16, VGPR0=K2,VGPR1=K3.
// B layout (4x16): lane%16 = N, lane/16 selects K pair, VGPRs stripe K in pair.
// D layout: VGPR r -> M = r + 8*(lane>=16), N = lane%16.
// ---------------------------------------------------------------------------
__global__ void __launch_bounds__(32)
mlp_gemm_wmma(const float* __restrict__ A, const float* __restrict__ Bm,
              const float* __restrict__ bias, float* __restrict__ out,
              int K, int do_relu) {
  const int lane  = threadIdx.x;
  const int m     = lane & 15;
  const int klane = (lane >> 4) * 2;
  const int n     = blockIdx.x * 16 + (lane & 15);

  v8f acc = {};
  for (int k0 = 0; k0 < K; k0 += 4) {
    v2f a, b;
    a.x = A[m * K + k0 + klane];
    a.y = A[m * K + k0 + klane + 1];
    b.x = Bm[(size_t)(k0 + klane) * CKK + n];
    b.y = Bm[(size_t)(k0 + klane + 1) * CKK + n];
    acc = __builtin_amdgcn_wmma_f32_16x16x4_f32(
        /*neg_a=*/false, a, /*neg_b=*/false, b,
        /*c_mod=*/(short)0, acc, /*reuse_a=*/false, /*reuse_b=*/false);
  }

  const float bv = bias[n];
  const int mbase = (lane >> 4) * 8;
#pragma unroll
  for (int r = 0; r < 8; ++r) {
    float v = acc[r] + bv;
    if (do_relu) v = v > 0.0f ? v : 0.0f;
    out[(size_t)(mbase + r) * CKK + n] = v;
  }
}

// ---------------------------------------------------------------------------
// Row softmax over 576 logits per batch row. One block per row.
// ---------------------------------------------------------------------------
__global__ void __launch_bounds__(256)
softmax_rows(const float* __restrict__ logits, float* __restrict__ kw,
             float* __restrict__ out_tail) {
  __shared__ float red[256];
  const int row = blockIdx.x;
  const int tid = threadIdx.x;
  const float* rp = logits + (size_t)row * CKK;

  float m = -INFINITY;
  for (int i = tid; i < CKK; i += 256) m = fmaxf(m, rp[i]);
  red[tid] = m;
  __syncthreads();
  for (int s = 128; s > 0; s >>= 1) {
    if (tid < s) red[tid] = fmaxf(red[tid], red[tid + s]);
    __syncthreads();
  }
  const float rmax = red[0];
  __syncthreads();

  float sum = 0.0f;
  for (int i = tid; i < CKK; i += 256) sum += __expf(rp[i] - rmax);
  red[tid] = sum;
  __syncthreads();
  for (int s = 128; s > 0; s >>= 1) {
    if (tid < s) red[tid] += red[tid + s];
    __syncthreads();
  }
  const float inv = 1.0f / red[0];

  for (int i = tid; i < CKK; i += 256) {
    float v = __expf(rp[i] - rmax) * inv;
    kw[(size_t)row * CKK + i]       = v;
    out_tail[(size_t)row * CKK + i] = v;
  }
}

// ---------------------------------------------------------------------------
// Depthwise dynamic 3x3 conv with reflect padding.
// Block = 256 threads handles an 8-row strip of one full-width 256-col plane.
// 10 halo rows staged into LDS with 16B-granularity transfers:
//   - async GLOBAL_LOAD_ASYNC_TO_LDS_B128 (memory->LDS direct) when available,
//   - float4 load/store fallback otherwise.
// Output written with non-temporal stores (write-once stream, > L2 capacity).
// ---------------------------------------------------------------------------
__global__ void __launch_bounds__(256)
dynconv(const float* __restrict__ Fd, const float* __restrict__ kw,
        float* __restrict__ Fout) {
  __shared__ __align__(16) float tile[(TILE_H + 2) * WW];
  __shared__ float wsh[9];

  const int tid   = threadIdx.x;
  const int bid   = blockIdx.x;
  const int plane = bid >> 5;           // 1024 planes (b*64 + c)
  const int y0    = (bid & 31) * TILE_H;

  const float* src = Fd + (size_t)plane * (HH * WW);
  float* dst       = Fout + (size_t)plane * (HH * WW);

  if (tid < 9) wsh[tid] = kw[(size_t)plane * 9 + tid];

  // Stage (TILE_H+2) x 256 floats = 640 float4 chunks into LDS.
  for (int i = tid; i < (TILE_H + 2) * (WW / 4); i += 256) {
    const int r  = i >> 6;              // row within tile (0..9)
    const int c4 = (i & 63) * 4;        // column (multiple of 4)
    int y = y0 - 1 + r;
    y = (y < 0) ? -y : ((y > HH - 1) ? (2 * (HH - 1) - y) : y);
    const float* gp = src + (size_t)y * WW + c4;
#if HAVE_ASYNC_LDS
    __builtin_amdgcn_global_load_async_to_lds_b128(
        (GLOBAL_AS v4i_vs*)(float*)gp,
        (LDS_AS v4i_vs*)&tile[r * WW + c4],
        /*offset=*/0, /*cpol=*/0);
#else
    const float4 v = *reinterpret_cast<const float4*>(gp);
    *reinterpret_cast<float4*>(&tile[r * WW + c4]) = v;
#endif
  }
#if HAVE_ASYNC_LDS
#if __has_builtin(__builtin_amdgcn_s_wait_asynccnt)
  __builtin_amdgcn_s_wait_asynccnt(0);
#else
  asm volatile("s_wait_asynccnt 0x0" ::: "memory");
#endif
#endif
  __syncthreads();

  const int xm1 = (tid == 0) ? 1 : tid - 1;            // reflect left edge
  const int xp1 = (tid == WW - 1) ? WW - 2 : tid + 1;  // reflect right edge
  const float w0 = wsh[0], w1 = wsh[1], w2 = wsh[2];
  const float w3 = wsh[3], w4 = wsh[4], w5 = wsh[5];
  const float w6 = wsh[6], w7 = wsh[7], w8 = wsh[8];

#pragma unroll
  for (int yy = 0; yy < TILE_H; ++yy) {
    const float* r0 = &tile[yy * WW];
    const float* r1 = r0 + WW;
    const float* r2 = r1 + WW;
    float acc;
    acc  = w0 * r0[xm1] + w1 * r0[tid] + w2 * r0[xp1];
    acc += w3 * r1[xm1] + w4 * r1[tid] + w5 * r1[xp1];
    acc += w6 * r2[xm1] + w7 * r2[tid] + w8 * r2[xp1];
    __builtin_nontemporal_store(acc, &dst[(size_t)(y0 + yy) * WW + tid]);
  }
}

extern "C" void kernel_launch(void* const* d_in, const int* in_sizes, int n_in,
                              void* d_out, int out_size, void* d_ws, size_t ws_size,
                              hipStream_t stream) {
  (void)in_sizes; (void)n_in; (void)out_size; (void)ws_size;

  const float* Fd = (const float*)d_in[0];   // (16,64,256,256)
  const float* Fc = (const float*)d_in[1];   // (16,512)
  const float* W1 = (const float*)d_in[2];   // (512,576)
  const float* b1 = (const float*)d_in[3];   // (576,)
  const float* W2 = (const float*)d_in[4];   // (576,576)
  const float* b2 = (const float*)d_in[5];   // (576,)

  float* out    = (float*)d_out;
  float* Fout   = out;                                        // 16*64*256*256
  float* kw_out = out + (size_t)BATCH * CIN * HH * WW;        // 16*64*3*3

  float* h      = (float*)d_ws;          // 16*576
  float* logits = h + BATCH * CKK;       // 16*576
  float* kwbuf  = logits + BATCH * CKK;  // 16*576

  // Layer 1: h = relu(Fc @ W1 + b1)     (K = 512)
  mlp_gemm_wmma<<<CKK / 16, 32, 0, stream>>>(Fc, W1, b1, h, FCD, 1);
  // Layer 2: logits = h @ W2 + b2       (K = 576)
  mlp_gemm_wmma<<<CKK / 16, 32, 0, stream>>>(h, W2, b2, logits, CKK, 0);
  // Softmax -> kernel weights (workspace + d_out tail)
  softmax_rows<<<BATCH, 256, 0, stream>>>(logits, kwbuf, kw_out);
  // Depthwise dynamic conv
  dynconv<<<BATCH * CIN * (HH / TILE_H), 256, 0, stream>>>(Fd, kwbuf, Fout);
}